// DetectionLoss_32100585570364
// MI455X (gfx1250) — compile-verified
//
#include <hip/hip_runtime.h>
#include <math.h>

// Detection loss (3-scale). Only cells hit by a target contribute to any loss
// term (everything is masked by pos_mask), so we compute the loss exactly via
// a sparse gather over <= 64*60 cells per scale instead of dense passes over
// ~95MB of predictions. Deterministic: fixed-order reductions, no atomics.
//
// CDNA5 paths used:
//  - GLOBAL_LOAD_ASYNC_TO_LDS_B128/B32 (ASYNCcnt) to stage the per-batch
//    target table directly into LDS without a VGPR round-trip.
//  - s_wait_asynccnt 0 before the LDS dedup scan.
//  - global_prefetch_b8 (__builtin_prefetch) to overlap the scattered
//    channel gathers with the 60-iteration dedup scan.

namespace {
constexpr int kNC   = 6;
constexpr int kT    = 60;   // targets per batch
constexpr int kB    = 64;   // batch
constexpr int kCh   = 11;   // channels per prediction map
constexpr float kEps = 1e-7f;

__device__ __forceinline__ float iou_xywh(float px, float py, float pw, float ph,
                                          float tx, float ty, float tw, float th) {
    // Follow reference formula literally (xyxy conversion, clip, eps).
    float x11 = px - pw * 0.5f, y11 = py - ph * 0.5f;
    float x12 = px + pw * 0.5f, y12 = py + ph * 0.5f;
    float x21 = tx - tw * 0.5f, y21 = ty - th * 0.5f;
    float x22 = tx + tw * 0.5f, y22 = ty + th * 0.5f;
    float iw = fmaxf(fminf(x12, x22) - fmaxf(x11, x21), 0.0f);
    float ih = fmaxf(fminf(y12, y22) - fmaxf(y11, y21), 0.0f);
    float inter = iw * ih;
    float a1 = (x12 - x11) * (y12 - y11);
    float a2 = (x22 - x21) * (y22 - y21);
    return inter / (a1 + a2 - inter + kEps);
}

// LDS byte offset of a __shared__ object: generic pointers to LDS carry the
// LDS offset in their low 32 bits (ISA 10.2 aperture mapping).
__device__ __forceinline__ unsigned lds_offset(const void* p) {
    return (unsigned)(uintptr_t)p;
}
} // namespace

// One block per (scale, batch). 64 threads = 2 wave32 waves; threads 0..59
// each own one target.
__global__ __launch_bounds__(64)
void det_loss_scale_kernel(const float* __restrict__ p3,
                           const float* __restrict__ p4,
                           const float* __restrict__ p5,
                           const int*   __restrict__ tcls,
                           const float* __restrict__ tbox,
                           float*       __restrict__ partials /* [3*kB][4] */) {
    const int blk = blockIdx.x;          // 0..191
    const int s   = blk / kB;            // scale index
    const int b   = blk - s * kB;        // batch index

    const float* pred; int H, W;
    if (s == 0)      { pred = p3; H = 160; W = 160; }
    else if (s == 1) { pred = p4; H =  80; W =  80; }
    else             { pred = p5; H =  40; W =  40; }

    __shared__ int   s_cell[kT];
    __shared__ __align__(16) float s_box[kT][4];   // 16B rows -> b128-aligned
    __shared__ int   s_cls[kT];
    __shared__ float s_acc[4][64];

    const int t = threadIdx.x;

    // Stage this batch's target table straight into LDS via async copies.
    if (t < kT) {
        const float* gbox = tbox + ((size_t)b * kT + t) * 4;   // 16B-aligned row
        const int*   gcls = tcls + (size_t)b * kT + t;
        unsigned lbox = lds_offset(&s_box[t][0]);
        unsigned lcls = lds_offset(&s_cls[t]);
        asm volatile("global_load_async_to_lds_b128 %0, %1, off"
                     :: "v"(lbox), "v"(gbox) : "memory");
        asm volatile("global_load_async_to_lds_b32 %0, %1, off"
                     :: "v"(lcls), "v"(gcls) : "memory");
    }
    asm volatile("s_wait_asynccnt 0x0" ::: "memory");   // own wave's copies done
    __syncthreads();                                    // other wave's copies done

    float tx = 0.f, ty = 0.f, tw = 0.f, th = 0.f;
    int gx = 0, gy = 0;
    if (t < kT) {
        tx = s_box[t][0]; ty = s_box[t][1]; tw = s_box[t][2]; th = s_box[t][3];
        gx = min(max((int)(tx * (float)W), 0), W - 1);   // trunc-toward-zero == astype(int32)
        gy = min(max((int)(ty * (float)H), 0), H - 1);
        s_cell[t] = gy * W + gx;
    }
    __syncthreads();

    float npos = 0.f, cls_sum = 0.f, iou_l = 0.f, inner_l = 0.f;
    if (t < kT) {
        const size_t cs   = (size_t)H * W;                       // channel stride
        const float* base = pred + (size_t)b * kCh * cs + (size_t)gy * W + gx;

        // gfx1250 global_prefetch_b8: start pulling the 11 scattered channel
        // cachelines while we run the 60-iteration LDS dedup scan below.
        #pragma unroll
        for (int c = 0; c < kCh; ++c)
            __builtin_prefetch(base + (size_t)c * cs, 0, 3);

        const int cell = s_cell[t];
        bool winner   = true;     // true iff t is the highest index mapping to this cell
        unsigned cmsk = 0u;       // union of classes scattered into this cell
        for (int u = 0; u < kT; ++u) {
            if (s_cell[u] == cell) {
                cmsk |= 1u << (s_cls[u] & 31);
                winner = winner && (u <= t);
            }
        }

        if (winner) {
            npos = 1.f;
            // Class BCE over channels 0..5 at this cell.
            #pragma unroll
            for (int c = 0; c < kNC; ++c) {
                float p  = base[(size_t)c * cs];
                float tv = (float)((cmsk >> c) & 1u);
                cls_sum += fmaxf(p, 0.f) - p * tv + log1pf(expf(-fabsf(p)));
            }
            // Box channels 7..10 (channel 6 = objectness, unused by this loss).
            float px = base[7 * cs], py = base[8 * cs];
            float pw = base[9 * cs], ph = base[10 * cs];
            iou_l   = 1.f - iou_xywh(px, py, pw, ph, tx, ty, tw, th);
            inner_l = 1.f - iou_xywh(px, py, pw * 0.7f, ph * 0.7f,
                                     tx, ty, tw * 0.7f, th * 0.7f);
        }
    }

    s_acc[0][t] = npos;
    s_acc[1][t] = cls_sum;
    s_acc[2][t] = iou_l;
    s_acc[3][t] = inner_l;
    __syncthreads();

    if (t == 0) {   // fixed-order block reduction -> deterministic across replays
        float a0 = 0.f, a1 = 0.f, a2 = 0.f, a3 = 0.f;
        for (int u = 0; u < 64; ++u) {
            a0 += s_acc[0][u]; a1 += s_acc[1][u];
            a2 += s_acc[2][u]; a3 += s_acc[3][u];
        }
        float* out = partials + (size_t)blk * 4;
        out[0] = a0; out[1] = a1; out[2] = a2; out[3] = a3;
    }
}

__global__ void det_loss_finalize_kernel(const float* __restrict__ partials,
                                         float* __restrict__ out) {
    if (threadIdx.x != 0 || blockIdx.x != 0) return;
    float cls_total = 0.f, box_total = 0.f;
    for (int s = 0; s < 3; ++s) {
        float np = 0.f, cl = 0.f, io = 0.f, in = 0.f;
        for (int b = 0; b < kB; ++b) {          // fixed order -> deterministic
            const float* p = partials + (size_t)(s * kB + b) * 4;
            np += p[0]; cl += p[1]; io += p[2]; in += p[3];
        }
        float inv     = 1.f / (np + 1e-8f);
        float cls_s   = cl * inv;
        float iou_t   = io * inv;
        float inner_t = in * inv;
        // inner_iou = (1-INNER_W)*iou + INNER_W*inner; box = 0.5*iou + 0.5*inner_iou
        float inner_iou = 0.5f * iou_t + 0.5f * inner_t;
        float box_s     = 0.5f * iou_t + 0.5f * inner_iou;
        cls_total += cls_s;
        box_total += box_s;
    }
    cls_total *= (1.0f / 3.0f);
    box_total *= (1.0f / 3.0f);
    out[0] = 0.5f * cls_total + 7.5f * box_total;   // CLS_W, BOX_W
    out[1] = cls_total;
    out[2] = box_total;
}

extern "C" void kernel_launch(void* const* d_in, const int* in_sizes, int n_in,
                              void* d_out, int out_size, void* d_ws, size_t ws_size,
                              hipStream_t stream) {
    (void)in_sizes; (void)n_in; (void)out_size; (void)ws_size;
    const float* p3   = (const float*)d_in[0];   // (64,11,160,160)
    const float* p4   = (const float*)d_in[1];   // (64,11,80,80)
    const float* p5   = (const float*)d_in[2];   // (64,11,40,40)
    const int*   tcls = (const int*)  d_in[3];   // (64,60)
    const float* tbox = (const float*)d_in[4];   // (64,60,4)

    float* partials = (float*)d_ws;              // 192*4 floats = 3 KB, fully rewritten each call

    det_loss_scale_kernel<<<3 * kB, 64, 0, stream>>>(p3, p4, p5, tcls, tbox, partials);
    det_loss_finalize_kernel<<<1, 32, 0, stream>>>(partials, (float*)d_out);
}